// DynamicFilterLayer_13357348290903
// MI455X (gfx1250) — compile-verified
//
#include <hip/hip_runtime.h>

typedef float v2f __attribute__((ext_vector_type(2)));
typedef float v8f __attribute__((ext_vector_type(8)));

#define KH 11
#define KW 11
#define NTAP 121
#define H 256
#define W 256
#define NB 16
#define PADR 5
#define TILE_ROWS 12     // 11 image rows + 1 zero row (not used by taps 0..120 except row guard)
#define TILE_PITCH 144   // >= 138 needed cols; IDXC(t+2)-IDXC(t) is +2 or +135
#define WG_PIX 128       // 8 waves * 16 pixels
#define HW (H * W)

// compile-time LDS tile index for tap t (row-major, pitch TILE_PITCH)
#define IDXC(t) (((t) / KW) * TILE_PITCH + ((t) % KW))

__global__ __launch_bounds__(256)
void dynfilter_wmma_kernel(const float* __restrict__ image,
                           const float* __restrict__ filters,
                           float* __restrict__ out)
{
    __shared__ float tile[TILE_ROWS * TILE_PITCH];

    const int bb  = blockIdx.z;
    const int y   = blockIdx.y;
    const int x0  = blockIdx.x * WG_PIX;
    const int tid = threadIdx.x;

    // ---- cooperative stage of the image tile: rows y-5..y+5, cols x0-5.. ----
    const float* imgb = image + (size_t)bb * HW;
    #pragma unroll
    for (int k = 0; k < (TILE_ROWS * TILE_PITCH + 255) / 256; ++k) {
        const int i = tid + k * 256;
        if (i < TILE_ROWS * TILE_PITCH) {
            const int r  = i / TILE_PITCH;
            const int c  = i % TILE_PITCH;
            const int yr = y + r - PADR;
            const int xc = x0 + c - PADR;
            float v = 0.0f;
            if (r < KH && (unsigned)yr < (unsigned)H && (unsigned)xc < (unsigned)W)
                v = imgb[yr * W + xc];
            tile[i] = v;
        }
    }
    __syncthreads();

    const int wave   = tid >> 5;     // 0..7
    const int lane   = tid & 31;
    const int half   = lane >> 4;    // A: K+0/1 vs K+2/3 halves; B: same
    const int lanelo = lane & 15;    // M (patch row) / N (filter col) id
    const int woff   = wave * 16;
    const int xw     = x0 + woff;

    // Per-lane loop-invariant A bases: tap t+2 sits at +2 (same row) or +135 (row wrap)
    const float* tA2   = &tile[woff + lanelo + 2 * half];
    const float* tA135 = &tile[woff + lanelo + 135 * half];

    // Per-lane loop-invariant B base: filters[b, 2*half, y, xw+lanelo]
    const float* fptr = filters + (((size_t)bb * NTAP + 2 * half) * H + y) * W + xw + lanelo;

    v8f acc = {};

    // taps 0..119: 30 chunks of K=4, all addresses compile-time immediates
    #pragma unroll
    for (int kc = 0; kc < 30; ++kc) {
        const int t0 = 4 * kc;       // component-0 tap (for half=0); half=1 adds +2 via base
        const int t1 = t0 + 1;

        v2f a;
        a.x = (((t0 % KW) <= KW - 3) ? tA2 : tA135)[IDXC(t0)];
        a.y = (((t1 % KW) <= KW - 3) ? tA2 : tA135)[IDXC(t1)];

        v2f bm;
        bm.x = __builtin_nontemporal_load(fptr + (size_t)t0 * HW);
        bm.y = __builtin_nontemporal_load(fptr + (size_t)t1 * HW);

        acc = __builtin_amdgcn_wmma_f32_16x16x4_f32(
            false, a, false, bm, (short)0, acc, false, false);
    }

    // ---- extract diag: p<8 -> lane p, acc[p]; p>=8 -> lane p+16, acc[p-8] ----
    const bool lo = (lane < 8);
    const bool hi = (lane >= 24);
    const int  sel = lo ? lane : (lane - 24);
    float r = 0.0f;
    #pragma unroll
    for (int j = 0; j < 8; ++j)
        r = (sel == j) ? acc[j] : r;

    if (lo || hi) {
        const int pix = lo ? lane : (lane - 16);
        // tap 120 (dy=+5, dx=+5) handled as a plain FMA on the extracted scalar
        const float a120 = tile[IDXC(120) + woff + pix];
        const float f120 = __builtin_nontemporal_load(
            filters + (((size_t)bb * NTAP + 120) * H + y) * W + xw + pix);
        r += a120 * f120;

        float* op = out + ((size_t)bb * H + y) * W + xw + pix;
        __builtin_nontemporal_store(r, op);
    }
}

extern "C" void kernel_launch(void* const* d_in, const int* in_sizes, int n_in,
                              void* d_out, int out_size, void* d_ws, size_t ws_size,
                              hipStream_t stream) {
    const float* image   = (const float*)d_in[0];   // (16,1,256,256) f32
    const float* filters = (const float*)d_in[1];   // (16,121,256,256) f32
    float* out = (float*)d_out;                     // (16,1,256,256) f32

    dim3 grid(W / WG_PIX, H, NB);   // (2, 256, 16)
    dim3 block(256);
    dynfilter_wmma_kernel<<<grid, block, 0, stream>>>(image, filters, out);
}